// GAE_55216099558155
// MI455X (gfx1250) — compile-verified
//
#include <hip/hip_runtime.h>
#include <math.h>

#define N_NODES 12288
#define N_EDGES 393216
#define IN_DIM  1433
#define H1      32
#define H2      16

typedef __attribute__((ext_vector_type(2))) float v2f;
typedef __attribute__((ext_vector_type(8))) float v8f;

// D = A(16x4, f32) * B(4x16, f32) + C(16x16, f32), full-precision f32 WMMA.
__device__ __forceinline__ v8f wmma_f32_4(v2f a, v2f b, v8f c) {
    return __builtin_amdgcn_wmma_f32_16x16x4_f32(
        /*neg_a=*/false, a, /*neg_b=*/false, b,
        /*c_mod=*/(short)0, c, /*reuse_a=*/false, /*reuse_b=*/false);
}

// sigmoid with hardware transcendentals only: v_exp_f32 + v_rcp_f32
// (avoids the ~10-op IEEE division expansion in the 151M-element decode loop)
__device__ __forceinline__ float fast_sigmoid(float x) {
    return __builtin_amdgcn_rcpf(1.f + __expf(-x));
}

// ---------------------------------------------------------------------------
// GEMM: C[M x (NT*16)] = A[M x K] @ B[K x (NT*16)], row-major.
// One wave per 16-row stripe computes ALL NT column tiles: the A fragment is
// loaded once per k-step and feeds NT WMMAs (halves A traffic for NT=2).
// RELU_A applies relu to A on load (fuses the activation of the previous
// layer). STREAM_A uses non-temporal loads for streamed-once A (the 70 MB X).
// ---------------------------------------------------------------------------
template <int NT, bool RELU_A, bool STREAM_A>
__global__ void gemm_wmma_f32(const float* __restrict__ A,
                              const float* __restrict__ B,
                              float* __restrict__ C,
                              int M, int K) {
    const int lane = threadIdx.x & 31;
    const int wave = (blockIdx.x * blockDim.x + threadIdx.x) >> 5;
    const int mtiles = M >> 4;
    if (wave >= mtiles) return;            // wave-uniform: EXEC stays all-1s

    const int ti = wave;
    const int l15 = lane & 15;
    const int koff = (lane >> 4) << 1;     // lanes 0-15 -> K{0,1}; 16-31 -> K{2,3}
    const int Nc = NT * 16;
    const float* arow = A + (long long)(ti * 16 + l15) * K;

    v8f acc[NT];
#pragma unroll
    for (int t = 0; t < NT; ++t) acc[t] = (v8f){};

    const int K4 = K & ~3;
    for (int kb = 0; kb < K4; kb += 4) {
        v2f a;
        if (STREAM_A) {
            a.x = __builtin_nontemporal_load(arow + kb + koff);
            a.y = __builtin_nontemporal_load(arow + kb + koff + 1);
        } else {
            a.x = arow[kb + koff];
            a.y = arow[kb + koff + 1];
        }
        if (RELU_A) { a.x = fmaxf(a.x, 0.f); a.y = fmaxf(a.y, 0.f); }
#pragma unroll
        for (int t = 0; t < NT; ++t) {
            v2f b;
            b.x = B[(kb + koff) * Nc + t * 16 + l15];
            b.y = B[(kb + koff + 1) * Nc + t * 16 + l15];
            acc[t] = wmma_f32_4(a, b, acc[t]);
        }
    }
    if (K4 < K) {                          // tail (K=1433 -> one leftover column)
        const int k0 = K4 + koff;
        v2f a;
        a.x = (k0     < K) ? arow[k0]     : 0.f;
        a.y = (k0 + 1 < K) ? arow[k0 + 1] : 0.f;
        if (RELU_A) { a.x = fmaxf(a.x, 0.f); a.y = fmaxf(a.y, 0.f); }
#pragma unroll
        for (int t = 0; t < NT; ++t) {
            v2f b;
            b.x = (k0     < K) ? B[k0 * Nc + t * 16 + l15]       : 0.f;
            b.y = (k0 + 1 < K) ? B[(k0 + 1) * Nc + t * 16 + l15] : 0.f;
            acc[t] = wmma_f32_4(a, b, acc[t]);
        }
    }

    // C/D layout: VGPR r holds row (r + 8*(lane>=16)), column = lane&15
    const int mbase = ti * 16 + ((lane >> 4) << 3);
#pragma unroll
    for (int t = 0; t < NT; ++t)
#pragma unroll
        for (int r = 0; r < 8; ++r)
            C[(long long)(mbase + r) * Nc + t * 16 + l15] = acc[t][r];
}

// ---------------------------------------------------------------------------
// COO SpMM scatter: out[row[e], j] += val[e] * in[col[e], j]
// One lane per (edge, feature). DIM is 32 or 16 (power of two).
// ---------------------------------------------------------------------------
template <int DIM>
__global__ void spmm_scatter(const float* __restrict__ vin,
                             const float* __restrict__ ev,
                             const int* __restrict__ er,
                             const int* __restrict__ ec,
                             float* __restrict__ vout) {
    const int t = blockIdx.x * blockDim.x + threadIdx.x;
    const int e = t >> (DIM == 32 ? 5 : 4);
    const int j = t & (DIM - 1);
    if (e >= N_EDGES) return;
    const float v = ev[e] * vin[ec[e] * DIM + j];
    unsafeAtomicAdd(&vout[er[e] * DIM + j], v);  // global_atomic_add_f32
}

__global__ void zero_kernel(float* __restrict__ p, int n) {
    const int i = blockIdx.x * blockDim.x + threadIdx.x;
    if (i < n) p[i] = 0.f;
}

// ---------------------------------------------------------------------------
// Decode: A_pred = sigmoid(Z @ Z^T), Z is [N x 16] row-major.
// One wave per 16x16 tile; K=16 -> 4 f32 WMMAs. Both fragments are aligned
// float2 loads from Z (row stride 16 floats = 64 B). A_pred (604 MB) is
// written with non-temporal stores: it is produced once, never re-read.
// ---------------------------------------------------------------------------
__global__ void decode_sigmoid(const float* __restrict__ Z,
                               float* __restrict__ Ap, int Nn) {
    const int lane = threadIdx.x & 31;
    const int wv = threadIdx.x >> 5;
    const int tj = blockIdx.x * (blockDim.x >> 5) + wv;
    const int ti = blockIdx.y;
    const int ntiles = Nn >> 4;
    if (tj >= ntiles) return;

    const int l15 = lane & 15;
    const int koff = (lane >> 4) << 1;
    const float* zA = Z + (long long)(ti * 16 + l15) * H2;  // A rows
    const float* zB = Z + (long long)(tj * 16 + l15) * H2;  // B = Z^T columns

    v8f acc = {};
#pragma unroll
    for (int kb = 0; kb < H2; kb += 4) {
        v2f a = *(const v2f*)(zA + kb + koff);
        v2f b = *(const v2f*)(zB + kb + koff);
        acc = wmma_f32_4(a, b, acc);
    }

    const int col = tj * 16 + l15;
    const int mbase = ti * 16 + ((lane >> 4) << 3);
#pragma unroll
    for (int r = 0; r < 8; ++r) {
        const float s = fast_sigmoid(acc[r]);
        __builtin_nontemporal_store(s, &Ap[(long long)(mbase + r) * Nn + col]);
    }
}

// ---------------------------------------------------------------------------
extern "C" void kernel_launch(void* const* d_in, const int* in_sizes, int n_in,
                              void* d_out, int out_size, void* d_ws, size_t ws_size,
                              hipStream_t stream) {
    const float* X  = (const float*)d_in[0];
    const float* W1 = (const float*)d_in[1];
    const float* W2 = (const float*)d_in[2];
    const float* W3 = (const float*)d_in[3];
    const float* ev = (const float*)d_in[4];
    const int*   er = (const int*)d_in[5];
    const int*   ec = (const int*)d_in[6];

    float* Apred = (float*)d_out;                              // [N, N]
    float* Z     = Apred + (size_t)N_NODES * N_NODES;          // [N, 16]

    float* ws = (float*)d_ws;
    float* T1 = ws;                    // X@W1           [N, 32]
    float* S1 = T1 + N_NODES * H1;     // spmm accum     [N, 32] (relu fused into next gemm)
    float* T2 = S1 + N_NODES * H1;     // relu(S1)@W2    [N, 32]
    float* S2 = T2 + N_NODES * H1;     // spmm accum     [N, 32]
    float* T3 = S2 + N_NODES * H1;     // relu(S2)@W3    [N, 16]

    const int B = 256;
    auto blocks = [](int n, int b) { return (n + b - 1) / b; };
    const int gemm_blk = ((N_NODES >> 4) * 32 + B - 1) / B;    // 1 wave / 16-row stripe

    // zero scatter accumulators (d_ws / d_out are poisoned, not re-zeroed)
    zero_kernel<<<blocks(N_NODES * H1, B), B, 0, stream>>>(S1, N_NODES * H1);
    zero_kernel<<<blocks(N_NODES * H1, B), B, 0, stream>>>(S2, N_NODES * H1);
    zero_kernel<<<blocks(N_NODES * H2, B), B, 0, stream>>>(Z,  N_NODES * H2);

    // layer 1: S1 = spmm(X @ W1)          (X streamed non-temporally, read once)
    gemm_wmma_f32<2, false, true><<<gemm_blk, B, 0, stream>>>(X, W1, T1, N_NODES, IN_DIM);
    spmm_scatter<32><<<blocks(N_EDGES * 32, B), B, 0, stream>>>(T1, ev, er, ec, S1);

    // layer 2: S2 = spmm(relu(S1) @ W2)   (relu fused into A-load)
    gemm_wmma_f32<2, true, false><<<gemm_blk, B, 0, stream>>>(S1, W2, T2, N_NODES, H1);
    spmm_scatter<32><<<blocks(N_EDGES * 32, B), B, 0, stream>>>(T2, ev, er, ec, S2);

    // layer 3: Z = spmm(relu(S2) @ W3)
    gemm_wmma_f32<1, true, false><<<gemm_blk, B, 0, stream>>>(S2, W3, T3, N_NODES, H1);
    spmm_scatter<16><<<blocks(N_EDGES * 16, B), B, 0, stream>>>(T3, ev, er, ec, Z);

    // decode: A_pred = sigmoid(Z @ Z^T)   (non-temporal 604 MB store)
    dim3 dgrid((N_NODES >> 4) / 8, N_NODES >> 4);   // 96 x 768, 8 waves/block
    decode_sigmoid<<<dgrid, B, 0, stream>>>(Z, Apred, N_NODES);
}